// Orthogonal_64493228917335
// MI455X (gfx1250) — compile-verified
//
#include <hip/hip_runtime.h>
#include <hip/hip_bf16.h>

typedef __attribute__((ext_vector_type(2))) float v2f;
typedef __attribute__((ext_vector_type(8))) float v8f;

#define D 256
#define PITCH 260   // LDS row pitch in floats: stride%64 == 4 -> conflict-free A reads

// ---------------------------------------------------------------------------
// Kernel 1: c[t] = 2 / dot(U[t], U[t])
// ---------------------------------------------------------------------------
__global__ void Orthogonal_norms_kernel(const float* __restrict__ U,
                                        float* __restrict__ c) {
    int t = threadIdx.x;  // 256 threads, 1 block
    const float* row = U + (size_t)t * D;
    float s = 0.f;
    #pragma unroll 8
    for (int i = 0; i < D; ++i) s = fmaf(row[i], row[i], s);
    c[t] = 2.0f / s;
}

// ---------------------------------------------------------------------------
// Kernel 2: compose the 256 Householder reflections into Mt = M^T, where
//   M = H(U[255]) * H(U[254]) * ... * H(U[0]),  Xout = X * M.
// Column j of M = apply H(U[0]) first, ..., H(U[255]) last to e_j.
// One wave (32 lanes) per column; column held in 8 VGPRs per lane.
// Mt[j][i] = M[i][j]  (row j of Mt is column j of M) -> contiguous stores,
// and contiguous float2 B-fragment loads in the GEMM.
// ---------------------------------------------------------------------------
__global__ void Orthogonal_compose_kernel(const float* __restrict__ U,
                                          const float* __restrict__ c,
                                          float* __restrict__ Mt) {
    const int lane = threadIdx.x & 31;
    const int j    = blockIdx.x * 8 + (threadIdx.x >> 5);  // column of M

    float a[8];
    #pragma unroll
    for (int m = 0; m < 8; ++m) a[m] = (lane + 32 * m == j) ? 1.0f : 0.0f;

    for (int t = 0; t < D; ++t) {
        const float* v = U + (size_t)t * D;
        float vv[8];
        float p = 0.f;
        #pragma unroll
        for (int m = 0; m < 8; ++m) {
            vv[m] = v[lane + 32 * m];          // coalesced: 32 consecutive floats
            p = fmaf(vv[m], a[m], p);
        }
        // wave32 reduction
        #pragma unroll
        for (int off = 16; off > 0; off >>= 1) p += __shfl_xor(p, off, 32);
        const float coef = c[t] * p;
        #pragma unroll
        for (int m = 0; m < 8; ++m) a[m] = fmaf(-coef, vv[m], a[m]);
    }

    float* dst = Mt + (size_t)j * D;
    #pragma unroll
    for (int m = 0; m < 8; ++m) dst[lane + 32 * m] = a[m];
}

// ---------------------------------------------------------------------------
// Kernel 3: Xout = X @ M using v_wmma_f32_16x16x4_f32 (fp32 WMMA, wave32).
// Workgroup = 256 threads = 8 waves, covers 32 rows x 256 cols of output.
// Wave w: row-tile (w>>2), col-tiles (w&3)*4 .. +3  -> 4 v8f accumulators.
// A fragment (16x4): lane l(0-15): rows, VGPR0/1 = K,K+1; lanes 16-31: K+2,K+3.
// B fragment (4x16): lane l: col = l%16, VGPR0/1 = Mt[col][K + 2*(l>>4) + 0/1].
// ---------------------------------------------------------------------------
__global__ void __launch_bounds__(256)
Orthogonal_gemm_kernel(const float* __restrict__ X,
                       const float* __restrict__ Mt,
                       float* __restrict__ out) {
    __shared__ float Xs[32 * PITCH];

    const int tid = threadIdx.x;
    // Stage contiguous 32x256 strip of X (32 KB) into LDS, padded pitch.
    const float* strip = X + (size_t)blockIdx.x * 32 * D;
    #pragma unroll
    for (int jj = 0; jj < 8; ++jj) {
        int fidx = (tid + jj * 256) * 4;            // float index in strip
        int row = fidx >> 8;                        // /256
        int col = fidx & 255;
        float4 val = *(const float4*)(strip + fidx);
        *(float4*)(&Xs[row * PITCH + col]) = val;
    }
    __syncthreads();

    const int lane = tid & 31;
    const int w    = tid >> 5;     // wave 0..7
    const int rt   = w >> 2;       // row tile 0..1
    const int cg   = w & 3;        // col group 0..3 (each = 4 col tiles)
    const int l16  = lane & 15;
    const int kh   = (lane >> 4) << 1;  // 0 or 2

    const float* arow  = &Xs[(rt * 16 + l16) * PITCH + kh];
    const int    cbase = cg * 64;

    v8f acc[4] = {};

    #pragma unroll 4
    for (int k = 0; k < D; k += 4) {
        float2 a2 = *(const float2*)(arow + k);
        v2f av = {a2.x, a2.y};
        #pragma unroll
        for (int t = 0; t < 4; ++t) {
            const float* bptr = Mt + (size_t)(cbase + t * 16 + l16) * D + k + kh;
            float2 b2 = *(const float2*)bptr;
            v2f bv = {b2.x, b2.y};
            acc[t] = __builtin_amdgcn_wmma_f32_16x16x4_f32(
                false, av, false, bv, (short)0, acc[t], false, false);
        }
    }

    // C/D layout: lane l, VGPR r -> row = r + 8*(l>>4), col = l%16
    const size_t rbase = (size_t)blockIdx.x * 32 + rt * 16 + (lane >> 4) * 8;
    #pragma unroll
    for (int t = 0; t < 4; ++t) {
        const int col = cbase + t * 16 + l16;
        #pragma unroll
        for (int r = 0; r < 8; ++r) {
            out[(rbase + r) * D + col] = acc[t][r];
        }
    }
}

// ---------------------------------------------------------------------------
extern "C" void kernel_launch(void* const* d_in, const int* in_sizes, int n_in,
                              void* d_out, int out_size, void* d_ws, size_t ws_size,
                              hipStream_t stream) {
    const float* X = (const float*)d_in[0];
    const float* U = (const float*)d_in[1];
    float* out = (float*)d_out;

    const int N = in_sizes[0] / D;  // 131072

    float* c  = (float*)d_ws;                       // 256 floats
    float* Mt = (float*)((char*)d_ws + 1024);       // 256*256 floats

    Orthogonal_norms_kernel<<<1, 256, 0, stream>>>(U, c);
    Orthogonal_compose_kernel<<<32, 256, 0, stream>>>(U, c, Mt);
    Orthogonal_gemm_kernel<<<N / 32, 256, 0, stream>>>(X, Mt, out);
}